// AttentionBasedLinkingModule_10471130268013
// MI455X (gfx1250) — compile-verified
//
#include <hip/hip_runtime.h>

typedef __attribute__((ext_vector_type(2))) float v2f;
typedef __attribute__((ext_vector_type(8))) float v8f;

#define D_MODEL 512
#define HIDDEN  500
#define NSPAN   256
#define KCON    1024
#define LMAX    64
#define LP1     65
#define PROWS   (NSPAN * LP1)   /* 16640, multiple of 16 */
#define NEG_INF_F (-1e30f)
#define LDS_STRIDE 516          /* 512 + 4 pad words: distinct LDS banks per lane */

#define WMMA_F32(A_, B_, C_) \
    __builtin_amdgcn_wmma_f32_16x16x4_f32(false, (A_), false, (B_), (short)0, (C_), false, false)

// ---------------------------------------------------------------------------
// One-time weight prep: transpose into N-major 512x512 with zero padding so
// every inner-loop B-fragment load is an unguarded contiguous b64.
//   WsT [n][k] = Ws[k][n]
//   W1aT[n][k] = W1[k    ][n]  (n<500 else 0)
//   W1bT[n][k] = W1[512+k][n]  (n<500 else 0)
//   W2T [n][k] = W2[k][n]      (n<500 && k<500 else 0)
// plus padded b1p/b2p/W3p (zeros past 500).
// ---------------------------------------------------------------------------
__global__ void prep_weights(const float* __restrict__ Ws, const float* __restrict__ W1,
                             const float* __restrict__ W2, const float* __restrict__ b1,
                             const float* __restrict__ b2, const float* __restrict__ W3,
                             float* __restrict__ WsT,  float* __restrict__ W1aT,
                             float* __restrict__ W1bT, float* __restrict__ W2T,
                             float* __restrict__ b1p,  float* __restrict__ b2p,
                             float* __restrict__ W3p)
{
    const int n = blockIdx.x;                        // output row (column of original)
    const bool nv = (n < HIDDEN);
    for (int k = threadIdx.x; k < D_MODEL; k += blockDim.x) {
        WsT [(size_t)n * D_MODEL + k] = Ws[(size_t)k * D_MODEL + n];
        W1aT[(size_t)n * D_MODEL + k] = nv ? W1[(size_t)k * HIDDEN + n] : 0.0f;
        W1bT[(size_t)n * D_MODEL + k] = nv ? W1[(size_t)(D_MODEL + k) * HIDDEN + n] : 0.0f;
        W2T [(size_t)n * D_MODEL + k] = (nv && k < HIDDEN) ? W2[(size_t)k * HIDDEN + n] : 0.0f;
    }
    if (threadIdx.x == 0) {
        b1p[n] = nv ? b1[n] : 0.0f;
        b2p[n] = nv ? b2[n] : 0.0f;
        W3p[n] = nv ? W3[n] : 0.0f;
    }
}

// ---------------------------------------------------------------------------
// fp32 WMMA GEMM, K = 512 fixed: C[m,n] = act(sum_k A[m,k]*Bt[n][k] + bias[n])
// A: M x 512 row-major. Bt: 512 x 512 transposed/zero-padded. C: M x 512.
// One 16x64 strip per wave (1 A-frag load -> 4 WMMAs per k-step).
// ---------------------------------------------------------------------------
__global__ void gemm_wmma_f32(const float* __restrict__ A, const float* __restrict__ Bt,
                              const float* __restrict__ bias, float* __restrict__ C,
                              int tM, int relu)
{
    const int wave = blockIdx.x * (blockDim.x >> 5) + (threadIdx.x >> 5);
    if (wave >= tM * 8) return;                      // 8 strips of 64 cols cover 512
    const int tm = wave % tM;
    const int tc = wave / tM;

    const int lane = threadIdx.x & 31;
    const int half = lane >> 4;                      // selects K pair (ISA 16x4 f32 layout)
    const int lm   = lane & 15;

    const float* __restrict__ Ap = A  + (size_t)((tm << 4) + lm) * D_MODEL;
    const float* __restrict__ Bp = Bt + (size_t)((tc << 6) + lm) * D_MODEL;

    v8f acc0 = {}, acc1 = {}, acc2 = {}, acc3 = {};
#pragma unroll 4
    for (int k = 0; k < D_MODEL; k += 4) {
        const int ka = k + (half << 1);
        v2f a  = *(const v2f*)(Ap + ka);
        v2f b0 = *(const v2f*)(Bp + ka);
        v2f b1v = *(const v2f*)(Bp + 16 * D_MODEL + ka);
        v2f b2v = *(const v2f*)(Bp + 32 * D_MODEL + ka);
        v2f b3v = *(const v2f*)(Bp + 48 * D_MODEL + ka);
        acc0 = WMMA_F32(a, b0,  acc0);
        acc1 = WMMA_F32(a, b1v, acc1);
        acc2 = WMMA_F32(a, b2v, acc2);
        acc3 = WMMA_F32(a, b3v, acc3);
    }

    v8f accs[4] = {acc0, acc1, acc2, acc3};
#pragma unroll
    for (int c = 0; c < 4; ++c) {
        const int bcol = (tc << 6) + (c << 4) + lm;
        const float bv = bias ? bias[bcol] : 0.0f;
#pragma unroll
        for (int r = 0; r < 8; ++r) {                // C layout: VGPR r -> row r (+8 half 1)
            float v = accs[c][r] + bv;
            if (relu) v = fmaxf(v, 0.0f);
            const int m = (tm << 4) + r + (half << 3);
            C[(size_t)m * D_MODEL + bcol] = v;
        }
    }
}

// ---------------------------------------------------------------------------
// Fused layer-1 gather/add/ReLU (to LDS) + layer-2 WMMA GEMM + W3 dot.
// Block = 256 threads (8 waves) owns 16 gathered rows p = m0..m0+15,
// p = n*65 + l (l==64 -> sentinel). Wave wv sweeps 64 cols (tiles 4wv..4wv+3).
// ---------------------------------------------------------------------------
__global__ void mlp2_score(const float* __restrict__ span_part,
                           const float* __restrict__ know_part,
                           const float* __restrict__ sent_part,
                           const float* __restrict__ b1p,
                           const int*   __restrict__ s2c,
                           const float* __restrict__ W2T,
                           const float* __restrict__ b2p,
                           const float* __restrict__ W3p,
                           const float* __restrict__ b3,
                           float* __restrict__ scores)
{
    __shared__ float sA[16][LDS_STRIDE];
    __shared__ float s_score[16];

    const int m0  = blockIdx.x << 4;
    const int tid = threadIdx.x;

    // --- build 16 x 512 ReLU'd layer-1 tile (cols >= 500 are exact zeros) ---
    {
        const int r  = tid >> 4;                     // 0..15
        const int c0 = tid & 15;
        const int p  = m0 + r;
        const int n  = p / LP1;
        const int l  = p - n * LP1;
        const float* __restrict__ src = (l == LMAX)
            ? (sent_part + (size_t)n * D_MODEL)
            : (know_part + (size_t)s2c[n * LMAX + l] * D_MODEL);
        const float* __restrict__ spn = span_part + (size_t)n * D_MODEL;
        for (int j = c0; j < D_MODEL; j += 16)       // padded cols of parts are 0; b1p is 0 there
            sA[r][j] = fmaxf(spn[j] + src[j] + b1p[j], 0.0f);
    }
    if (tid < 16) s_score[tid] = b3[0];
    __syncthreads();

    const int lane = tid & 31;
    const int wv   = tid >> 5;                       // 0..7 -> 64-column strip
    const int half = lane >> 4;
    const int lm   = lane & 15;
    const float* __restrict__ Ap = &sA[lm][0];
    const float* __restrict__ Bp = W2T + (size_t)((wv << 6) + lm) * D_MODEL;

    v8f acc0 = {}, acc1 = {}, acc2 = {}, acc3 = {};
#pragma unroll 4
    for (int k = 0; k < D_MODEL; k += 4) {
        const int ka = k + (half << 1);
        v2f a  = *(const v2f*)(Ap + ka);             // ds_load_b64, bank-conflict-free
        v2f b0 = *(const v2f*)(Bp + ka);
        v2f b1v = *(const v2f*)(Bp + 16 * D_MODEL + ka);
        v2f b2v = *(const v2f*)(Bp + 32 * D_MODEL + ka);
        v2f b3v = *(const v2f*)(Bp + 48 * D_MODEL + ka);
        acc0 = WMMA_F32(a, b0,  acc0);
        acc1 = WMMA_F32(a, b1v, acc1);
        acc2 = WMMA_F32(a, b2v, acc2);
        acc3 = WMMA_F32(a, b3v, acc3);
    }

    v8f accs[4] = {acc0, acc1, acc2, acc3};
    float w3v[4], b2v[4];
#pragma unroll
    for (int c = 0; c < 4; ++c) {
        const int bcol = (wv << 6) + (c << 4) + lm;  // padded cols: W3p/b2p = 0
        w3v[c] = W3p[bcol];
        b2v[c] = b2p[bcol];
    }
#pragma unroll
    for (int r = 0; r < 8; ++r) {
        float part = 0.0f;
#pragma unroll
        for (int c = 0; c < 4; ++c)
            part += fmaxf(accs[c][r] + b2v[c], 0.0f) * w3v[c];
        for (int off = 8; off > 0; off >>= 1)        // butterfly over 16 lanes per half
            part += __shfl_xor(part, off, 32);
        if (lm == 0) atomicAdd(&s_score[r + (half << 3)], part);
    }
    __syncthreads();
    if (tid < 16) scores[m0 + tid] = s_score[tid];
}

// ---------------------------------------------------------------------------
// Masked softmax over 65 scores per span + weighted embedding gather.
// Writes features (N x 512) then probs (N x 65) into d_out.
// ---------------------------------------------------------------------------
__global__ void softmax_feat(const float* __restrict__ scores,
                             const int*   __restrict__ s2c,
                             const int*   __restrict__ lengths,
                             const float* __restrict__ knowledge,
                             const float* __restrict__ sentinel,
                             float* __restrict__ out)
{
    __shared__ float sp[LP1];
    __shared__ int   sidx[LMAX];

    const int n   = blockIdx.x;
    const int tid = threadIdx.x;
    if (tid < LMAX) sidx[tid] = s2c[n * LMAX + tid];

    if (tid == 0) {
        int len = lengths[n]; if (len < 1) len = 1;
        float mx = NEG_INF_F;
        for (int l = 0; l < LP1; ++l) {
            float s = scores[n * LP1 + l];
            const bool valid = (l == LMAX) || (l < len);
            s = valid ? s : NEG_INF_F;
            sp[l] = s;
            mx = fmaxf(mx, s);
        }
        float sum = 0.0f;
        for (int l = 0; l < LP1; ++l) { float e = expf(sp[l] - mx); sp[l] = e; sum += e; }
        const float inv = 1.0f / sum;
        for (int l = 0; l < LP1; ++l) sp[l] *= inv;
    }
    __syncthreads();

    if (tid < LP1) out[(size_t)NSPAN * D_MODEL + n * LP1 + tid] = sp[tid];

    for (int j = tid; j < D_MODEL; j += blockDim.x) {
        float acc = 0.0f;
        for (int l = 0; l < LMAX; ++l)
            acc += sp[l] * knowledge[(size_t)sidx[l] * D_MODEL + j];
        acc += sp[LMAX] * sentinel[(size_t)n * D_MODEL + j];
        out[(size_t)n * D_MODEL + j] = acc;
    }
}

// ---------------------------------------------------------------------------
extern "C" void kernel_launch(void* const* d_in, const int* in_sizes, int n_in,
                              void* d_out, int out_size, void* d_ws, size_t ws_size,
                              hipStream_t stream)
{
    const float* span_embs = (const float*)d_in[0];   // 256 x 512
    const float* knowledge = (const float*)d_in[1];   // 1024 x 512
    const int*   s2c       = (const int*)  d_in[2];   // 256 x 64
    const int*   lengths   = (const int*)  d_in[3];   // 256
    const float* Ws        = (const float*)d_in[4];   // 512 x 512
    const float* bs        = (const float*)d_in[5];   // 512
    const float* W1        = (const float*)d_in[6];   // 1024 x 500
    const float* b1        = (const float*)d_in[7];   // 500
    const float* W2        = (const float*)d_in[8];   // 500 x 500
    const float* b2        = (const float*)d_in[9];   // 500
    const float* W3        = (const float*)d_in[10];  // 500 x 1
    const float* b3        = (const float*)d_in[11];  // 1

    float* ws = (float*)d_ws;
    const size_t MSZ = (size_t)D_MODEL * D_MODEL;     // 262144 floats
    float* WsT       = ws;
    float* W1aT      = WsT   + MSZ;
    float* W1bT      = W1aT  + MSZ;
    float* W2T       = W1bT  + MSZ;
    float* b1p       = W2T   + MSZ;
    float* b2p       = b1p   + D_MODEL;
    float* W3p       = b2p   + D_MODEL;
    float* sentinel  = W3p   + D_MODEL;               // 256*512
    float* span_part = sentinel  + (size_t)NSPAN * D_MODEL;
    float* know_part = span_part + (size_t)NSPAN * D_MODEL;  // 1024*512
    float* sent_part = know_part + (size_t)KCON  * D_MODEL;
    float* scores    = sent_part + (size_t)NSPAN * D_MODEL;  // 16640

    // one-time transpose/pad of all weights (L2-resident, ~4 MB)
    prep_weights<<<D_MODEL, 256, 0, stream>>>(Ws, W1, W2, b1, b2, W3,
                                              WsT, W1aT, W1bT, W2T, b1p, b2p, W3p);

    const dim3 blk(128);                              // 4 waves/block
    auto blocks_for = [](int tM) { return (tM * 8 + 3) / 4; };

    // sentinel = relu(span_embs @ Ws + bs)
    gemm_wmma_f32<<<blocks_for(NSPAN / 16), blk, 0, stream>>>(
        span_embs, WsT, bs, sentinel, NSPAN / 16, 1);
    // span_part = span_embs @ W1[:512]
    gemm_wmma_f32<<<blocks_for(NSPAN / 16), blk, 0, stream>>>(
        span_embs, W1aT, nullptr, span_part, NSPAN / 16, 0);
    // know_part = knowledge_embs @ W1[512:]
    gemm_wmma_f32<<<blocks_for(KCON / 16), blk, 0, stream>>>(
        knowledge, W1bT, nullptr, know_part, KCON / 16, 0);
    // sent_part = sentinel @ W1[512:]
    gemm_wmma_f32<<<blocks_for(NSPAN / 16), blk, 0, stream>>>(
        sentinel, W1bT, nullptr, sent_part, NSPAN / 16, 0);

    // fused gathered-row MLP layer 2 + scoring (16,640 of 262,400 rows)
    mlp2_score<<<PROWS / 16, 256, 0, stream>>>(
        span_part, know_part, sent_part, b1p, s2c, W2T, b2p, W3p, b3, scores);

    // masked softmax + attention-weighted embedding gather
    softmax_feat<<<NSPAN, 256, 0, stream>>>(
        scores, s2c, lengths, knowledge, sentinel, (float*)d_out);
}